// GCN_87144886436012
// MI455X (gfx1250) — compile-verified
//
#include <hip/hip_runtime.h>

// ---------------------------------------------------------------------------
// GCN (2x GCNConv + linear readout) for MI455X / gfx1250 (wave32, WMMA).
// GEMMs: V_WMMA_F32_16X16X4_F32, 2 M-tiles per wave, B tile staged transposed
// in LDS (ds_load_2addr_b64 per WMMA pair). Aggregation: one wave per edge,
// float4 gather per lane from L2-resident h, global_atomic_add_f32 scatter.
// ---------------------------------------------------------------------------

typedef float v2f __attribute__((ext_vector_type(2)));
typedef float v8f __attribute__((ext_vector_type(8)));

#define DHID 128
#define BSTRIDE 130   // LDS row stride in words: even (8B-aligned pairs) + bank spread

__global__ void k_init_deg(float* __restrict__ deg, int n) {
    int i = blockIdx.x * blockDim.x + threadIdx.x;
    if (i < n) deg[i] = 1.0f;            // self-loop contributes 1
}

__global__ void k_count_deg(const int* __restrict__ dst, float* __restrict__ deg, int e) {
    int i = blockIdx.x * blockDim.x + threadIdx.x;
    if (i < e) atomicAdd(&deg[dst[i]], 1.0f);
}

__global__ void k_rsqrt_inplace(float* __restrict__ deg, int n) {
    int i = blockIdx.x * blockDim.x + threadIdx.x;
    if (i < n) deg[i] = rsqrtf(deg[i]);
}

__global__ void k_zero4(float4* __restrict__ p, int n4) {
    int i = blockIdx.x * blockDim.x + threadIdx.x;
    if (i < n4) p[i] = make_float4(0.f, 0.f, 0.f, 0.f);
}

// C[M x ldc] = A[M x 128] * B[128 x ldb] (+ bias), valid output cols < nvalid.
// 256 threads = 8 waves; each wave computes TWO 16x16 tiles (32 rows) against
// a shared 128x16 B tile staged transposed in LDS.
__global__ void k_gemm_wmma_f32(const float* __restrict__ A,
                                const float* __restrict__ B,
                                const float* __restrict__ bias,
                                float* __restrict__ C,
                                int M, int ldb, int nvalid, int ldc,
                                int addbias) {
    const int K = DHID;
    __shared__ float Bt[16 * BSTRIDE];            // Bt[c][k], transposed B tile

    const int tn = blockIdx.y * 16;

    // ---- stage B tile (transposed, column clamped to nvalid-1) -------------
    for (int i = threadIdx.x; i < 16 * K; i += 256) {
        const int c = i & 15;                     // col within tile
        const int k = i >> 4;                     // k index
        int gc = tn + c; if (gc >= nvalid) gc = nvalid - 1;
        Bt[c * BSTRIDE + k] = B[(long)k * ldb + gc];
    }
    __syncthreads();

    const int lane  = threadIdx.x & 31;
    const int wave  = threadIdx.x >> 5;
    const int mrow  = lane & 15;
    const int khalf = lane >> 4;                  // 0: K pair {0,1}, 1: {2,3}
    const int col   = tn + mrow;
    const bool colok = (col < nvalid);

    // two M tiles per wave
    const int tm0 = (blockIdx.x * 16 + wave * 2) * 16;
    const int tm1 = tm0 + 16;
    if (tm0 >= M) return;                         // wave-uniform (after barrier)

    int ar0 = tm0 + mrow; if (ar0 >= M) ar0 = M - 1;
    int ar1 = tm1 + mrow; if (ar1 >= M) ar1 = M - 1;
    const float* arow0 = A + (long)ar0 * K;
    const float* arow1 = A + (long)ar1 * K;
    const float* brow  = &Bt[mrow * BSTRIDE];     // lane's column of B (by k)

    v8f acc0 = {};
    v8f acc1 = {};
#pragma unroll 4
    for (int k0 = 0; k0 < K; k0 += 4) {
        const int k = k0 + khalf * 2;
        v2f a0 = *(const v2f*)(arow0 + k);        // 2 consecutive K values / lane
        v2f a1 = *(const v2f*)(arow1 + k);
        v2f b  = *(const v2f*)(brow + k);         // ds_load_b64, shared by both
        acc0 = __builtin_amdgcn_wmma_f32_16x16x4_f32(
            false, a0, false, b, (short)0, acc0, false, false);
        acc1 = __builtin_amdgcn_wmma_f32_16x16x4_f32(
            false, a1, false, b, (short)0, acc1, false, false);
    }

    if (colok) {
        const float bv   = addbias ? bias[col] : 0.0f;
        const long  step = (long)ldc * sizeof(float);
        const int   rb0  = tm0 + khalf * 8;       // lanes16-31 hold rows M=8..15
        const int   rb1  = tm1 + khalf * 8;

        // tile 0: tm0 < M and tiles are 16-aligned vs M (M%16==0 here) -> whole
        // tile in range; keep a guarded fallback for the generic ragged case.
        if (tm0 + 16 <= M) {
            char* p = (char*)(C + (long)rb0 * ldc + col);
#pragma unroll
            for (int r = 0; r < 8; ++r) { *(float*)p = acc0[r] + bv; p += step; }
        } else {
#pragma unroll
            for (int r = 0; r < 8; ++r)
                if (rb0 + r < M) C[(long)(rb0 + r) * ldc + col] = acc0[r] + bv;
        }
        if (tm1 + 16 <= M) {
            char* p = (char*)(C + (long)rb1 * ldc + col);
#pragma unroll
            for (int r = 0; r < 8; ++r) { *(float*)p = acc1[r] + bv; p += step; }
        } else if (tm1 < M) {
#pragma unroll
            for (int r = 0; r < 8; ++r)
                if (rb1 + r < M) C[(long)(rb1 + r) * ldc + col] = acc1[r] + bv;
        }
    }
}

// out[dst] += h[src] * dinv[src]*dinv[dst]; edges e in [E, Etot) are the
// virtual self-loops (src = dst = e - E). One wave per edge, float4 per lane.
__global__ void k_aggregate(const float* __restrict__ h,
                            const int* __restrict__ src,
                            const int* __restrict__ dst,
                            const float* __restrict__ dinv,
                            float* __restrict__ out,
                            int E, int Etot) {
    const int wave = threadIdx.x >> 5;
    const int lane = threadIdx.x & 31;
    const int e = blockIdx.x * 8 + wave;
    if (e >= Etot) return;
    int s, d;
    if (e < E) { s = src[e]; d = dst[e]; }
    else       { s = e - E;  d = s;      }
    const float w = dinv[s] * dinv[d];
    const float4 hv = *(const float4*)(h + (long)s * DHID + lane * 4);
    float* orow = out + (long)d * DHID + lane * 4;
    atomicAdd(orow + 0, hv.x * w);
    atomicAdd(orow + 1, hv.y * w);
    atomicAdd(orow + 2, hv.z * w);
    atomicAdd(orow + 3, hv.w * w);
}

__global__ void k_bias_act(float* __restrict__ x, const float* __restrict__ b,
                           int n, int relu) {
    int i = blockIdx.x * blockDim.x + threadIdx.x;
    if (i < n) {
        float v = x[i] + b[i & (DHID - 1)];
        if (relu) v = fmaxf(v, 0.0f);
        x[i] = v;
    }
}

extern "C" void kernel_launch(void* const* d_in, const int* in_sizes, int n_in,
                              void* d_out, int out_size, void* d_ws, size_t ws_size,
                              hipStream_t stream) {
    const float* x    = (const float*)d_in[0];
    const int*   ei   = (const int*)  d_in[1];
    const float* W1   = (const float*)d_in[2];
    const float* b1   = (const float*)d_in[3];
    const float* W2   = (const float*)d_in[4];
    const float* b2   = (const float*)d_in[5];
    const float* Wout = (const float*)d_in[6];
    const float* bout = (const float*)d_in[7];

    const int N    = in_sizes[0] / DHID;      // 100000
    const int E    = in_sizes[1] / 2;         // 1600000
    const int DOUT = in_sizes[7];             // 40
    const int Etot = E + N;

    const int* srcI = ei;
    const int* dstI = ei + E;

    // workspace: dinv[N] | bufA[N*128] | bufB[N*128]
    char*  ws   = (char*)d_ws;
    float* dinv = (float*)ws;
    size_t off  = ((size_t)N * sizeof(float) + 255) & ~(size_t)255;
    float* bufA = (float*)(ws + off);
    float* bufB = (float*)(ws + off + (size_t)N * DHID * sizeof(float));

    const int BT = 256;
    const int mtiles  = (N + 15) / 16;                 // 16-row tiles
    const int gx_gemm = (mtiles + 15) / 16;            // 2 tiles/wave * 8 waves
    const int nElem   = N * DHID;
    const int n4      = nElem / 4;

    // --- degrees -> dinv -----------------------------------------------------
    k_init_deg<<<(N + BT - 1) / BT, BT, 0, stream>>>(dinv, N);
    k_count_deg<<<(E + BT - 1) / BT, BT, 0, stream>>>(dstI, dinv, E);
    k_rsqrt_inplace<<<(N + BT - 1) / BT, BT, 0, stream>>>(dinv, N);

    // --- layer 1: h = x @ W1 ; agg ; +b1 ; relu ------------------------------
    k_gemm_wmma_f32<<<dim3(gx_gemm, DHID / 16), BT, 0, stream>>>(
        x, W1, nullptr, bufA, N, DHID, DHID, DHID, 0);
    k_zero4<<<(n4 + BT - 1) / BT, BT, 0, stream>>>((float4*)bufB, n4);
    k_aggregate<<<(Etot + 7) / 8, BT, 0, stream>>>(bufA, srcI, dstI, dinv, bufB, E, Etot);
    k_bias_act<<<(nElem + BT - 1) / BT, BT, 0, stream>>>(bufB, b1, nElem, 1);

    // --- layer 2: h = h @ W2 ; agg ; +b2 -------------------------------------
    k_gemm_wmma_f32<<<dim3(gx_gemm, DHID / 16), BT, 0, stream>>>(
        bufB, W2, nullptr, bufA, N, DHID, DHID, DHID, 0);
    k_zero4<<<(n4 + BT - 1) / BT, BT, 0, stream>>>((float4*)bufB, n4);
    k_aggregate<<<(Etot + 7) / 8, BT, 0, stream>>>(bufA, srcI, dstI, dinv, bufB, E, Etot);
    k_bias_act<<<(nElem + BT - 1) / BT, BT, 0, stream>>>(bufB, b2, nElem, 0);

    // --- readout: out = h @ Wout + bout --------------------------------------
    k_gemm_wmma_f32<<<dim3(gx_gemm, (DOUT + 15) / 16), BT, 0, stream>>>(
        bufB, Wout, bout, (float*)d_out, N, DOUT, DOUT, DOUT, 1);
}